// MultiScaleTimeAwareSelfAttention_67491116089878
// MI455X (gfx1250) — compile-verified
//
#include <hip/hip_runtime.h>

#define DV   192
#define SEQ  4096
#define BS   8
#define D1   64
#define DVP  196   // padded LDS row stride (floats): 196 % 64 = 4 -> conflict-free b64 reads

typedef float v2f __attribute__((ext_vector_type(2)));
typedef float v8f __attribute__((ext_vector_type(8)));

// ---------------------------------------------------------------------------
// Kernel 1: fused projection GEMM  v[b,s,0:192] = x[b,s,:] @ [w1|w2|w3] + [b1|b2|b3]
// 256 threads = 8 waves per block; each wave owns one 16-row M tile.
// Each 192x64 weight chunk is DMA'd (GLOBAL_LOAD_ASYNC_TO_LDS_B32, ASYNCcnt)
// transposed into LDS (~50 KB) and shared by all 8 waves.
// WMMA: D(16x16,f32) = A(16x4 of x) * B(4x16 of W) + C.
// ---------------------------------------------------------------------------
__global__ __launch_bounds__(256)
void msta_proj_kernel(const float* __restrict__ x,
                      const float* __restrict__ w1, const float* __restrict__ b1,
                      const float* __restrict__ w2, const float* __restrict__ b2,
                      const float* __restrict__ w3, const float* __restrict__ b3,
                      float* __restrict__ v) {
    __shared__ float lwt[D1 * DVP];   // transposed weight chunk: lwt[col*DVP + row]
    __shared__ float lb[D1];

    const int tid  = threadIdx.x;
    const int wave = tid >> 5;
    const int lane = tid & 31;
    const int lm   = lane & 15;       // A row / B,C,D column within tile
    const int hi   = lane >> 4;       // selects K pair {0,1} vs {2,3}

    const int m0 = (blockIdx.x * 8 + wave) * 16;   // global row base of this wave

    const float* wptr[3] = {w1, w2, w3};
    const float* bptr[3] = {b1, b2, b3};

#pragma unroll
    for (int c = 0; c < 3; ++c) {
        // Stage chunk c transposed into LDS via CDNA5 async global->LDS DMA:
        //   lwt[col][row] = w_c[row][col]
        // Global side is fully coalesced (consecutive lanes -> consecutive
        // addresses); LDS side is per-lane scattered (VDST carries the LDS
        // byte address; HW adds LDS_BASE).  Tracked with ASYNCcnt.
        const float* w = wptr[c];
        for (int idx = tid; idx < DV * D1; idx += 256) {
            int row = idx >> 6;       // 0..191 (input dim)
            int col = idx & 63;       // 0..63  (output dim)
            unsigned lds_off = (unsigned)(size_t)(&lwt[col * DVP + row]);
            const float* gsrc = w + idx;
            asm volatile("global_load_async_to_lds_b32 %0, %1, off"
                         :: "v"(lds_off), "v"(gsrc) : "memory");
        }
        if (tid < D1) lb[tid] = bptr[c][tid];
        asm volatile("s_wait_asynccnt 0x0" ::: "memory");
        __syncthreads();

#pragma unroll
        for (int nn = 0; nn < 4; ++nn) {
            const int cb = nn * 16;                 // chunk-local column base
            v8f acc;
            float bias = lb[cb + lm];               // C/D col depends only on lane&15
#pragma unroll
            for (int r = 0; r < 8; ++r) acc[r] = bias;

            // A layout (16x4 f32): lanes 0-15 rows m0+lm, K = kk+{0,1};
            //                      lanes 16-31 same rows,  K = kk+{2,3}  -> one b64 load
            const float* arow = x   + (size_t)(m0 + lm) * DV + 2 * hi;
            // B layout (4x16 f32): lane col = cb+lm, VGPR0=K kk+2hi, VGPR1=K kk+2hi+1
            const float* brow = lwt + (cb + lm) * DVP + 2 * hi;

            for (int kk = 0; kk < DV; kk += 4) {
                v2f a = *(const v2f*)(arow + kk);
                v2f b = *(const v2f*)(brow + kk);
                acc = __builtin_amdgcn_wmma_f32_16x16x4_f32(
                        false, a, false, b, (short)0, acc, false, false);
            }
            // D layout: VGPR r -> row m0 + r + 8*hi, col = c*64 + cb + lm
            float* o = v + (size_t)(m0 + 8 * hi) * DV + c * D1 + cb + lm;
#pragma unroll
            for (int r = 0; r < 8; ++r) o[(size_t)r * DV] = acc[r];
        }
        __syncthreads();
    }
}

// ---------------------------------------------------------------------------
// Kernel 2: banded Toeplitz attention.
// out[b,i,sc*64+ch] = sum_j exp(-c_sc|i-j|)/Z(i) * v[b,j,sc*64+ch]
// Attention entries are generated on the fly (never touch memory); closed-form
// row normalizer Z(i).  One wave per (16-row tile, scale); 4 channel subtiles.
// ---------------------------------------------------------------------------
__global__ __launch_bounds__(256)
void msta_attn_kernel(const float* __restrict__ v, float* __restrict__ out) {
    const int tid  = threadIdx.x;
    const int wave = tid >> 5;
    const int lane = tid & 31;
    const int lm   = lane & 15;
    const int hi   = lane >> 4;

    const int wid = blockIdx.x * 8 + wave;   // 0..6143
    const int sc  = wid >> 11;               // scale 0..2  (2048 m-tiles each)
    const int mt  = wid & 2047;
    const int b   = mt >> 8;                 // SEQ/16 = 256 tiles per batch
    const int i0  = (mt & 255) << 4;         // sequence-local row base

    const float cdec[3]  = {5.0f, 1.0f, 0.2f};   // decay constants
    const int   dband[3] = {16, 32, 128};        // fp32-safe half-bandwidths
    const float c = cdec[sc];
    const int   D = dband[sc];

    // Z(i) = [ (1 - r^{i+1}) + (r - r^{S-i}) ] / (1 - r),  r = e^{-c}
    const float i_row = (float)(i0 + lm);        // this lane's absolute row
    const float r     = __expf(-c);
    const float Z     = ((1.0f - __expf(-c * (i_row + 1.0f))) +
                         (r    - __expf(-c * ((float)SEQ - i_row)))) / (1.0f - r);
    const float zinv  = 1.0f / Z;

    const float* vb = v + (size_t)b * SEQ * DV;

    v8f acc[4];
#pragma unroll
    for (int nn = 0; nn < 4; ++nn)
#pragma unroll
        for (int rr = 0; rr < 8; ++rr) acc[nn][rr] = 0.0f;

    for (int j0 = i0 - D; j0 <= i0 + D; j0 += 16) {   // wave-uniform band loop
        if (j0 < 0 || j0 >= SEQ) continue;            // 16-aligned: no partial tiles
        // Pull next j-tile's V rows toward the WGP while this tile computes
        if (j0 + 16 < SEQ)
            __builtin_prefetch(vb + (size_t)(j0 + 16 + 2 * hi) * DV + sc * D1 + lm, 0, 1);
#pragma unroll
        for (int kk = 0; kk < 16; kk += 4) {
            const int j = j0 + kk + 2 * hi;
            // A fragment: rows i_row, cols {j, j+1}; Toeplitz value on the fly
            v2f a;
            a[0] = __expf(-c * fabsf(i_row - (float)j))        * zinv;
            a[1] = __expf(-c * fabsf(i_row - (float)(j + 1)))  * zinv;
            // B fragment: V rows {j, j+1}, channel col sc*64 + nn*16 + lm
            const float* vj = vb + (size_t)j * DV + sc * D1 + lm;
#pragma unroll
            for (int nn = 0; nn < 4; ++nn) {
                v2f bf;
                bf[0] = vj[nn * 16];
                bf[1] = vj[nn * 16 + DV];
                acc[nn] = __builtin_amdgcn_wmma_f32_16x16x4_f32(
                            false, a, false, bf, (short)0, acc[nn], false, false);
            }
        }
    }

#pragma unroll
    for (int nn = 0; nn < 4; ++nn) {
        float* o = out + ((size_t)(b * SEQ) + i0 + 8 * hi) * DV + sc * D1 + nn * 16 + lm;
#pragma unroll
        for (int rr = 0; rr < 8; ++rr) o[(size_t)rr * DV] = acc[nn][rr];
    }
}

// ---------------------------------------------------------------------------
extern "C" void kernel_launch(void* const* d_in, const int* in_sizes, int n_in,
                              void* d_out, int out_size, void* d_ws, size_t ws_size,
                              hipStream_t stream) {
    const float* x  = (const float*)d_in[0];
    const float* w1 = (const float*)d_in[1];
    const float* b1 = (const float*)d_in[2];
    const float* w2 = (const float*)d_in[3];
    const float* b2 = (const float*)d_in[4];
    const float* w3 = (const float*)d_in[5];
    const float* b3 = (const float*)d_in[6];
    float* out = (float*)d_out;
    float* v   = (float*)d_ws;    // BS*SEQ*DV fp32 = 25.2 MB intermediate (L2-resident)

    // 32768 rows / (8 waves * 16 rows) = 256 blocks
    msta_proj_kernel<<<256, 256, 0, stream>>>(x, w1, b1, w2, b2, w3, b3, v);
    // 2048 m-tiles * 3 scales = 6144 wave tasks / 8 = 768 blocks
    msta_attn_kernel<<<768, 256, 0, stream>>>(v, out);
}